// ArithmeticCircuitGNN_57329223467299
// MI455X (gfx1250) — compile-verified
//
#include <hip/hip_runtime.h>

// GNN: 3x GCNConv (N=100000, E=1.6M, D=128) + LayerNorm/ReLU/residual.
// Memory-bound on edge scatter (L2-resident: 51.2MB feature matrix << 192MB L2).
// GEMMs use V_WMMA_F32_16X16X4_F32 (full fp32, matches reference precision).
// Scatter uses hardware global_atomic_add_f32 at device scope (inline asm).

#define D 128
#define LN_EPS 1e-5f
#define WSTRIDE 132   // LDS row stride in floats (132 % 64 == 4 -> conflict-free)

typedef __attribute__((ext_vector_type(2))) float v2f;
typedef __attribute__((ext_vector_type(8))) float v8f;

__device__ __forceinline__ void atomic_fadd(float* p, float v) {
  // Non-returning fp32 atomic add, device scope -> L2 atomic unit.
  asm volatile("global_atomic_add_f32 %0, %1, off scope:SCOPE_DEV"
               : : "v"(p), "v"(v) : "memory");
}

// ---------------- degree / normalization ----------------
__global__ void k_fill1(float* __restrict__ deg, int n) {
  int i = blockIdx.x * blockDim.x + threadIdx.x;
  if (i < n) deg[i] = 1.0f;  // self-loop
}

__global__ void k_deg(const int* __restrict__ dst, float* __restrict__ deg, int e) {
  int i = blockIdx.x * blockDim.x + threadIdx.x;
  if (i < e) atomic_fadd(&deg[dst[i]], 1.0f);  // integer-valued fp32 adds: exact
}

__global__ void k_rsqrt(float* __restrict__ dinv, int n) {
  int i = blockIdx.x * blockDim.x + threadIdx.x;
  if (i < n) dinv[i] = rsqrtf(dinv[i]);  // deg >= 1 always
}

// ---------------- WMMA GEMM: H = A @ W ; P = bias + H * dinv^2 (fused self-loop init) ----
// Block: 256 threads = 8 waves. Block handles 16 rows x 128 cols; wave w -> cols [16w,16w+16).
__global__ __launch_bounds__(256) void k_gemm_fused(
    const float* __restrict__ A, const float* __restrict__ W,
    const float* __restrict__ bias, const float* __restrict__ dinv,
    float* __restrict__ H, float* __restrict__ P, int n) {
  __shared__ float Alds[16 * WSTRIDE];
  __shared__ float Wlds[32 * WSTRIDE];

  const int t    = threadIdx.x;
  const int lane = t & 31;
  const int wave = t >> 5;
  const int row0 = blockIdx.x * 16;
  const int col0 = wave * 16;
  const bool hi  = lane >= 16;
  const int mn   = lane & 15;        // A-row / B-col / C-col index within tile
  const int koff = hi ? 2 : 0;       // K sub-offset per ISA A/B layout

  // Stage A tile (16 x 128) into LDS, coalesced
  for (int i = t; i < 16 * D; i += 256) {
    int r = i >> 7, c = i & (D - 1);
    int gr = row0 + r; if (gr >= n) gr = n - 1;
    Alds[r * WSTRIDE + c] = A[(size_t)gr * D + c];
  }

  v8f acc = {0.f, 0.f, 0.f, 0.f, 0.f, 0.f, 0.f, 0.f};

  for (int kc = 0; kc < 4; ++kc) {              // K chunks of 32
    __syncthreads();
    for (int i = t; i < 32 * D; i += 256) {     // stage W rows [kc*32, kc*32+32)
      int r = i >> 7, c = i & (D - 1);
      Wlds[r * WSTRIDE + c] = W[(size_t)(kc * 32 + r) * D + c];
    }
    __syncthreads();
#pragma unroll
    for (int kk = 0; kk < 8; ++kk) {            // 8 x (16x16x4) WMMA per chunk
      const int kl = kk * 4 + koff;
      v2f a, b;
      a.x = Alds[mn * WSTRIDE + kc * 32 + kl];
      a.y = Alds[mn * WSTRIDE + kc * 32 + kl + 1];
      b.x = Wlds[kl * WSTRIDE + col0 + mn];
      b.y = Wlds[(kl + 1) * WSTRIDE + col0 + mn];
      acc = __builtin_amdgcn_wmma_f32_16x16x4_f32(
          /*neg_a=*/false, a, /*neg_b=*/false, b,
          /*c_mod=*/(short)0, acc, /*reuse_a=*/false, /*reuse_b=*/false);
    }
  }

  // C/D layout: VGPR r -> row (r + hi*8), col = col0 + mn
  const int col = col0 + mn;
  const float bc = bias[col];
  const int rbase = row0 + (hi ? 8 : 0);
  if (row0 + 16 <= n) {
    // fast path: full tile in-bounds, straight-line stores (no divergence)
#pragma unroll
    for (int r = 0; r < 8; ++r) {
      const int grow = rbase + r;
      const float v  = acc[r];
      const float di = dinv[grow];
      H[(size_t)grow * D + col] = v;                 // for edge gather
      P[(size_t)grow * D + col] = bc + v * di * di;  // bias + self-loop term
    }
  } else {
#pragma unroll
    for (int r = 0; r < 8; ++r) {
      const int grow = rbase + r;
      if (grow < n) {
        const float v  = acc[r];
        const float di = dinv[grow];
        H[(size_t)grow * D + col] = v;
        P[(size_t)grow * D + col] = bc + v * di * di;
      }
    }
  }
}

// ---------------- edge scatter: out[dst] += H[src] * dinv[src]*dinv[dst] ----------------
// One wave per edge; each lane handles 4 consecutive features (128 = 32*4).
__global__ __launch_bounds__(256) void k_scatter(
    const float* __restrict__ H, const float* __restrict__ dinv,
    const int* __restrict__ src, const int* __restrict__ dst,
    float* __restrict__ out, int e) {
  int w = (int)(((long)blockIdx.x * 256 + threadIdx.x) >> 5);
  if (w >= e) return;
  w = __builtin_amdgcn_readfirstlane(w);   // provably wave-uniform -> scalar loads
  const int lane = threadIdx.x & 31;
  const int s = src[w], d = dst[w];
  const float sc = dinv[s] * dinv[d];
  const float4 h = *(const float4*)(H + (size_t)s * D + lane * 4);
  float* o = out + (size_t)d * D + lane * 4;
  atomic_fadd(o + 0, h.x * sc);
  atomic_fadd(o + 1, h.y * sc);
  atomic_fadd(o + 2, h.z * sc);
  atomic_fadd(o + 3, h.w * sc);
}

// ---------------- elementwise ----------------
__global__ void k_relu(const float* __restrict__ in, float* __restrict__ out, long n4) {
  long i = (long)blockIdx.x * blockDim.x + threadIdx.x;
  if (i >= n4) return;
  float4 v = ((const float4*)in)[i];
  float4 o = {fmaxf(v.x, 0.f), fmaxf(v.y, 0.f), fmaxf(v.z, 0.f), fmaxf(v.w, 0.f)};
  ((float4*)out)[i] = o;
}

__global__ void k_add_relu(const float* __restrict__ agg, float* __restrict__ ident, long n4) {
  long i = (long)blockIdx.x * blockDim.x + threadIdx.x;
  if (i >= n4) return;
  float4 v = ((const float4*)agg)[i];
  float4 id = ((float4*)ident)[i];
  float4 o = {fmaxf(v.x, 0.f) + id.x, fmaxf(v.y, 0.f) + id.y,
              fmaxf(v.z, 0.f) + id.z, fmaxf(v.w, 0.f) + id.w};
  ((float4*)ident)[i] = o;
}

// ---------------- LayerNorm: one wave per row (D=128 -> 4 floats/lane) ----------------
__global__ __launch_bounds__(256) void k_ln(
    const float* __restrict__ in, const float* __restrict__ g,
    const float* __restrict__ b, float* __restrict__ out, int n) {
  int row = (int)(((long)blockIdx.x * 256 + threadIdx.x) >> 5);
  if (row >= n) return;
  int lane = threadIdx.x & 31;
  const float4 v = *(const float4*)(in + (size_t)row * D + lane * 4);
  float s = v.x + v.y + v.z + v.w;
#pragma unroll
  for (int off = 16; off > 0; off >>= 1) s += __shfl_xor(s, off, 32);
  float mu = s * (1.0f / D);
  float dx = v.x - mu, dy = v.y - mu, dz = v.z - mu, dw = v.w - mu;
  float q = dx * dx + dy * dy + dz * dz + dw * dw;
#pragma unroll
  for (int off = 16; off > 0; off >>= 1) q += __shfl_xor(q, off, 32);
  float rs = rsqrtf(q * (1.0f / D) + LN_EPS);
  const float4 gg = *(const float4*)(g + lane * 4);
  const float4 bb = *(const float4*)(b + lane * 4);
  float4 o = {dx * rs * gg.x + bb.x, dy * rs * gg.y + bb.y,
              dz * rs * gg.z + bb.z, dw * rs * gg.w + bb.w};
  *(float4*)(out + (size_t)row * D + lane * 4) = o;
}

extern "C" void kernel_launch(void* const* d_in, const int* in_sizes, int n_in,
                              void* d_out, int out_size, void* d_ws, size_t ws_size,
                              hipStream_t stream) {
  const float* x    = (const float*)d_in[0];
  const int*   ei   = (const int*)d_in[1];   // [2, E] int32 (JAX x64 disabled)
  const float* W0   = (const float*)d_in[2];
  const float* b0   = (const float*)d_in[3];
  const float* W1   = (const float*)d_in[4];
  const float* b1   = (const float*)d_in[5];
  const float* W2   = (const float*)d_in[6];
  const float* b2   = (const float*)d_in[7];
  const float* ln_g = (const float*)d_in[8];
  const float* ln_b = (const float*)d_in[9];
  const float* fn_g = (const float*)d_in[10];
  const float* fn_b = (const float*)d_in[11];

  const int n = in_sizes[0] / D;
  const int e = in_sizes[1] / 2;
  const int* src = ei;
  const int* dst = ei + e;
  float* out = (float*)d_out;

  // Workspace: dinv[N] | t0[N*D] (GEMM out H) | t1[N*D] (aggregate) | t2[N*D] (identity)
  float* dinv = (float*)d_ws;
  size_t nAl = ((size_t)n + 255) & ~(size_t)255;
  float* t0 = dinv + nAl;
  float* t1 = t0 + (size_t)n * D;
  float* t2 = t1 + (size_t)n * D;

  const int  bn   = (n + 255) / 256;
  const int  be   = (e + 255) / 256;
  const int  gG   = (n + 15) / 16;             // GEMM blocks (16 rows each)
  const int  scG  = (e + 7) / 8;               // scatter: 8 waves (edges) / block
  const int  lnG  = (int)(((long)n * 32 + 255) / 256);
  const long nd4  = (long)n * D / 4;
  const int  ewG  = (int)((nd4 + 255) / 256);

  // degree -> dinv
  k_fill1 <<<bn, 256, 0, stream>>>(dinv, n);
  k_deg   <<<be, 256, 0, stream>>>(dst, dinv, e);
  k_rsqrt <<<bn, 256, 0, stream>>>(dinv, n);

  // layer 0: h = relu(gcn(x, W0, b0))
  k_gemm_fused<<<gG, 256, 0, stream>>>(x, W0, b0, dinv, t0, t1, n);
  k_scatter   <<<scG, 256, 0, stream>>>(t0, dinv, src, dst, t1, e);
  k_relu      <<<ewG, 256, 0, stream>>>(t1, t2, nd4);            // t2 = identity
  k_ln        <<<lnG, 256, 0, stream>>>(t2, ln_g, ln_b, out, n); // d_out as scratch

  // layer 1: h = relu(gcn(ln(h), W1, b1)) + identity
  k_gemm_fused<<<gG, 256, 0, stream>>>(out, W1, b1, dinv, t0, t1, n);
  k_scatter   <<<scG, 256, 0, stream>>>(t0, dinv, src, dst, t1, e);
  k_add_relu  <<<ewG, 256, 0, stream>>>(t1, t2, nd4);            // t2 = relu(t1)+t2

  // layer 2: out = ln(gcn(h, W2, b2))
  k_gemm_fused<<<gG, 256, 0, stream>>>(t2, W2, b2, dinv, t0, t1, n);
  k_scatter   <<<scG, 256, 0, stream>>>(t0, dinv, src, dst, t1, e);
  k_ln        <<<lnG, 256, 0, stream>>>(t1, fn_g, fn_b, out, n);
}